// CD_loss_75419625718565
// MI455X (gfx1250) — compile-verified
//
#include <hip/hip_runtime.h>
#include <hip/hip_bf16.h>

// ---------------------------------------------------------------------------
// Chamfer distance (KITTI depth -> point cloud) for MI455X / gfx1250.
// Pairwise sq-dist tiles computed with V_WMMA_F32_16X16X4_F32:
//   A row i = [x1,y1,z1,1], B col j = [-2x2,-2y2,-2z2, sq2_j (or 1e30 masked)]
//   tile(i,j) = sq2_j - 2*p1_i.p2_j ; add sq1_i after the row-min.
// Two passes (roles swapped) make both chamfer directions row-mins.
// ---------------------------------------------------------------------------

#define HREF 128
#define WREF 128
#define NPTS (HREF * WREF)      // 16384 points per cloud
#define NTILES (NPTS / 16)      // 1024 16-wide tiles
#define BIGF 1e30f

typedef float v2f __attribute__((ext_vector_type(2)));
typedef float v8f __attribute__((ext_vector_type(8)));

// ---------------------------------------------------------------- init ------
__global__ void cd_init_kernel(float* __restrict__ accum) {
    if (threadIdx.x < 4) accum[threadIdx.x] = 0.0f;
}

// ------------------------------------------------------- build points -------
// Per pixel: pinhole back-projection for target & pred, emit
//  A-format rows [x,y,z,1], B-format cols [-2x,-2y,-2z, sq | BIGF],
//  per-row squared norms, valid flag; count valid pixels.
__global__ __launch_bounds__(256) void cd_build_kernel(
    const float* __restrict__ pred,
    const float* __restrict__ target,
    const float* __restrict__ P,      // 3x4 row-major
    float* __restrict__ a_t, float* __restrict__ b_t,
    float* __restrict__ a_p, float* __restrict__ b_p,
    float* __restrict__ sq_t, float* __restrict__ sq_p,
    float* __restrict__ validf, float* __restrict__ accum)
{
    int p = blockIdx.x * blockDim.x + threadIdx.x;
    if (p >= NPTS) return;

    const float fu  = P[0],  cu  = P[2],  p03 = P[3];
    const float fv  = P[5],  cv  = P[6],  p13 = P[7];
    const float p23 = P[11];

    const float fx = (float)(p & (WREF - 1));
    const float fy = (float)(p >> 7);

    const float dt = target[p];
    const float dp = pred[p];
    const bool valid = dt > 0.0f;

    // target cloud (rows of D)
    float xt = (fx * (dt + p23) - (cu * dt + p03)) / fu;
    float yt = (fy * (dt + p23) - (cv * dt + p13)) / fv;
    float st = xt * xt + yt * yt + dt * dt;
    // pred cloud (cols of D)
    float xp = (fx * (dp + p23) - (cu * dp + p03)) / fu;
    float yp = (fy * (dp + p23) - (cv * dp + p13)) / fv;
    float sp = xp * xp + yp * yp + dp * dp;

    float4* A_t = (float4*)a_t;  float4* B_t = (float4*)b_t;
    float4* A_p = (float4*)a_p;  float4* B_p = (float4*)b_p;

    A_t[p] = make_float4(xt, yt, dt, 1.0f);
    A_p[p] = make_float4(xp, yp, dp, 1.0f);
    // Column masking (both directions use the target-derived valid mask):
    if (valid) {
        B_t[p] = make_float4(-2.0f * xt, -2.0f * yt, -2.0f * dt, st);
        B_p[p] = make_float4(-2.0f * xp, -2.0f * yp, -2.0f * dp, sp);
    } else {
        B_t[p] = make_float4(0.0f, 0.0f, 0.0f, BIGF);
        B_p[p] = make_float4(0.0f, 0.0f, 0.0f, BIGF);
    }
    sq_t[p]   = st;
    sq_p[p]   = sp;
    validf[p] = valid ? 1.0f : 0.0f;
    if (valid) atomicAdd(&accum[2], 1.0f);
}

// ------------------------------------------- one chamfer direction ----------
// One wave32 per 16-row A tile; streams all 1024 B tiles.
// Inner loop: global_load_b64 + v_wmma_f32_16x16x4_f32 + 8x v_min_num_f32.
__global__ __launch_bounds__(128) void cd_dist_kernel(
    const float* __restrict__ Apts,     // NPTS*4: rows [x,y,z,1]
    const float* __restrict__ Bpts,     // NPTS*4: cols [-2x,-2y,-2z, sq|BIGF]
    const float* __restrict__ sqrow,    // NPTS  : |row point|^2
    const float* __restrict__ validrow, // NPTS  : row mask
    float* __restrict__ accum, int slot)
{
    const int lane = threadIdx.x & 31;
    const int l15  = lane & 15;
    const int hi   = lane >> 4;                        // 0: K0/K1, 1: K2/K3
    const int wid  = blockIdx.x * (blockDim.x >> 5) + (threadIdx.x >> 5);

    // A 16x4 f32 layout (wave32): lane holds row M=l15; lo half K0,K1 / hi half K2,K3
    const v2f* A2 = (const v2f*)Apts;
    const v2f* B2 = (const v2f*)Bpts;
    const v2f a = A2[(wid * 16 + l15) * 2 + hi];

    v8f vmin;
#pragma unroll
    for (int r = 0; r < 8; ++r) vmin[r] = BIGF;

#pragma unroll 4
    for (int t = 0; t < NTILES; ++t) {
        // B 4x16 f32 layout mirrors A: lane holds col N=l15
        v2f b = B2[(t * 16 + l15) * 2 + hi];
        v8f c = {};
        c = __builtin_amdgcn_wmma_f32_16x16x4_f32(
                /*neg_a=*/false, a, /*neg_b=*/false, b,
                /*c_mod=*/(short)0, c, /*reuse_a=*/false, /*reuse_b=*/false);
#pragma unroll
        for (int r = 0; r < 8; ++r) vmin[r] = fminf(vmin[r], c[r]);
    }

    // Row-min across the 16 lane-columns inside each half-wave.
#pragma unroll
    for (int r = 0; r < 8; ++r) {
        float v = vmin[r];
#pragma unroll
        for (int off = 1; off < 16; off <<= 1)
            v = fminf(v, __shfl_xor(v, off, 32));
        vmin[r] = v;
    }

    // C/D layout: VGPR r -> row r (lanes 0-15) and row r+8 (lanes 16-31).
    // lane 0 finalizes rows base..base+7, lane 16 rows base+8..base+15.
    if (l15 == 0) {
        const int base = wid * 16 + hi * 8;
        float s = 0.0f;
#pragma unroll
        for (int r = 0; r < 8; ++r) {
            const int row = base + r;
            if (validrow[row] > 0.0f) s += sqrow[row] + vmin[r];
        }
        atomicAdd(&accum[slot], s);
    }
}

// --------------------------------------------------------------- finalize ---
__global__ void cd_final_kernel(const float* __restrict__ accum,
                                float* __restrict__ out) {
    if (threadIdx.x == 0)
        out[0] = (accum[0] + accum[1]) / accum[2];
}

// ---------------------------------------------------------------------------
extern "C" void kernel_launch(void* const* d_in, const int* in_sizes, int n_in,
                              void* d_out, int out_size, void* d_ws, size_t ws_size,
                              hipStream_t stream) {
    const float* pred   = (const float*)d_in[0];   // [1,1,128,128]
    const float* target = (const float*)d_in[1];   // [1,1,128,128]
    const float* P      = (const float*)d_in[2];   // [3,4]

    float* ws     = (float*)d_ws;
    float* a_t    = ws;                 // NPTS*4
    float* b_t    = a_t    + NPTS * 4;  // NPTS*4
    float* a_p    = b_t    + NPTS * 4;  // NPTS*4
    float* b_p    = a_p    + NPTS * 4;  // NPTS*4
    float* sq_t   = b_p    + NPTS * 4;  // NPTS
    float* sq_p   = sq_t   + NPTS;      // NPTS
    float* validf = sq_p   + NPTS;      // NPTS
    float* accum  = validf + NPTS;      // 4: {sum12, sum21, nvalid, pad}

    cd_init_kernel<<<1, 32, 0, stream>>>(accum);

    cd_build_kernel<<<NPTS / 256, 256, 0, stream>>>(
        pred, target, P, a_t, b_t, a_p, b_p, sq_t, sq_p, validf, accum);

    // dir 1-2: rows = target points, cols = pred points (masked by valid)
    cd_dist_kernel<<<NTILES / 4, 128, 0, stream>>>(a_t, b_p, sq_t, validf, accum, 0);
    // dir 2-1: rows = pred points, cols = target points (masked by valid)
    cd_dist_kernel<<<NTILES / 4, 128, 0, stream>>>(a_p, b_t, sq_p, validf, accum, 1);

    cd_final_kernel<<<1, 32, 0, stream>>>(accum, (float*)d_out);
}